// GCN_64561948393903
// MI455X (gfx1250) — compile-verified
//
#include <hip/hip_runtime.h>
#include <hip/hip_bf16.h>

typedef __attribute__((ext_vector_type(2))) float v2f;
typedef __attribute__((ext_vector_type(8))) float v8f;

// ---------- degree / norm precompute ----------

__global__ void k_fill1(float* p, int n) {
    int i = blockIdx.x * blockDim.x + threadIdx.x;
    if (i < n) p[i] = 1.0f;                      // self-loop contributes 1 to deg
}

__global__ void k_deg_count(const int* __restrict__ col, float* deg, int e) {
    int i = blockIdx.x * blockDim.x + threadIdx.x;
    if (i < e) atomicAdd(&deg[col[i]], 1.0f);
}

__global__ void k_dinv(float* deg, int n) {
    int i = blockIdx.x * blockDim.x + threadIdx.x;
    if (i < n) {
        float d = deg[i];
        deg[i] = (d > 0.0f) ? rsqrtf(d) : 0.0f;
    }
}

__global__ void k_norm(const int* __restrict__ row, const int* __restrict__ col,
                       const float* __restrict__ dinv, float* __restrict__ norm, int e) {
    int i = blockIdx.x * blockDim.x + threadIdx.x;
    if (i < e) norm[i] = dinv[row[i]] * dinv[col[i]];
}

// ---------- fp32 WMMA GEMM: out[M,Nout] = A[M,K] @ W[K,Nout] ----------
// One wave per 16x16 output tile. K must be a multiple of 4 (128/64/128 here).
// Nout may be < tile width (layer 3: Nout=2): columns are masked to zero.

__global__ __launch_bounds__(32) void k_gemm_wmma(
    const float* __restrict__ A, const float* __restrict__ W,
    float* __restrict__ out, int M, int K, int Nout, int ntilesN) {
    const int lane = threadIdx.x & 31;
    const int half = lane >> 4;     // 0: K={0,1} of quad, 1: K={2,3}
    const int mr   = lane & 15;     // row of A / col of B
    const int tm   = blockIdx.x / ntilesN;
    const int tn   = blockIdx.x - tm * ntilesN;

    int rowA = tm * 16 + mr;
    if (rowA >= M) rowA = M - 1;                 // clamp: keep loads in-bounds, EXEC full
    const int  colB   = tn * 16 + mr;
    const bool bvalid = (colB < Nout);
    const int  colBc  = bvalid ? colB : (Nout - 1);
    const float bmask = bvalid ? 1.0f : 0.0f;    // mask padded columns multiplicatively

    const float* __restrict__ Arow = A + (size_t)rowA * K;

    v8f acc = {};
    for (int k0 = 0; k0 < K; k0 += 4) {
        const int ka = k0 + half * 2;
        v2f a = *(const v2f*)(Arow + ka);        // A[m][ka], A[m][ka+1]
        v2f b;
        b[0] = W[(size_t)(ka + 0) * Nout + colBc] * bmask;
        b[1] = W[(size_t)(ka + 1) * Nout + colBc] * bmask;
        acc = __builtin_amdgcn_wmma_f32_16x16x4_f32(
            /*neg_a=*/false, a, /*neg_b=*/false, b,
            /*c_mod=*/(short)0, acc, /*reuse_a=*/false, /*reuse_b=*/false);
    }

    if (bvalid) {
#pragma unroll
        for (int r = 0; r < 8; ++r) {
            const int m = tm * 16 + r + 8 * half;   // C/D layout: vgpr r -> M = r + 8*half
            if (m < M) out[(size_t)m * Nout + colB] = acc[r];
        }
    }
}

// ---------- aggregation: out = D^-1/2 A D^-1/2 (lin), with self-loops ----------

__global__ void k_agg_init(const float* __restrict__ lin, const float* __restrict__ dinv,
                           float* __restrict__ agg, int total, int logh) {
    int t = blockIdx.x * blockDim.x + threadIdx.x;
    if (t < total) {
        int i = t >> logh;
        float d = dinv[i];
        agg[t] = d * d * lin[t];                 // self-loop term norm = dinv[i]^2
    }
}

__global__ void k_agg_edges(const int* __restrict__ row, const int* __restrict__ col,
                            const float* __restrict__ norm, const float* __restrict__ lin,
                            float* agg, int e, int logh) {
    unsigned t = blockIdx.x * blockDim.x + threadIdx.x;
    unsigned ei = t >> logh;
    unsigned j  = t & ((1u << logh) - 1u);
    if (ei < (unsigned)e) {
        int r = row[ei], c = col[ei];
        float v = norm[ei] * lin[((size_t)r << logh) + j];
        atomicAdd(&agg[((size_t)c << logh) + j], v);
    }
}

__global__ void k_bias_relu(float* agg, const float* __restrict__ bias, int total, int logh) {
    int t = blockIdx.x * blockDim.x + threadIdx.x;
    if (t < total) {
        int j = t & ((1 << logh) - 1);
        float v = agg[t] + bias[j];
        agg[t] = fmaxf(v, 0.0f);
    }
}

__global__ void k_bias_logsoftmax(const float* __restrict__ agg, const float* __restrict__ bias,
                                  float* __restrict__ out, int n) {
    int i = blockIdx.x * blockDim.x + threadIdx.x;
    if (i < n) {
        float v0 = agg[2 * i + 0] + bias[0];
        float v1 = agg[2 * i + 1] + bias[1];
        float m  = fmaxf(v0, v1);
        float lse = m + logf(expf(v0 - m) + expf(v1 - m));
        out[2 * i + 0] = v0 - lse;
        out[2 * i + 1] = v1 - lse;
    }
}

// ---------- driver ----------

extern "C" void kernel_launch(void* const* d_in, const int* in_sizes, int n_in,
                              void* d_out, int out_size, void* d_ws, size_t ws_size,
                              hipStream_t stream) {
    const float* x  = (const float*)d_in[0];
    const int*   ei = (const int*)d_in[1];
    const float* W1 = (const float*)d_in[2];
    const float* b1 = (const float*)d_in[3];
    const float* W2 = (const float*)d_in[4];
    const float* b2 = (const float*)d_in[5];
    const float* W3 = (const float*)d_in[6];
    const float* b3 = (const float*)d_in[7];

    const int P = 128, H = 64, C = 2;
    const int N = in_sizes[0] / P;
    const int E = in_sizes[1] / 2;
    const int* row = ei;        // edge_index[0] = source
    const int* col = ei + E;    // edge_index[1] = target

    float* wsf = (float*)d_ws;
    size_t off = 0;
    auto take = [&](size_t nf) {
        float* p = wsf + off;
        off += (nf + 63) & ~(size_t)63;
        return p;
    };
    float* dinv = take((size_t)N);          // deg -> dinv in place
    float* norm = take((size_t)E);
    float* lin  = take((size_t)N * 128);    // GEMM output (max width 128)
    float* aggA = take((size_t)N * 128);
    float* aggB = take((size_t)N * 128);

    const int T = 256;
    auto blk = [](long long n, int t) { return (unsigned)((n + t - 1) / t); };

    // norm precompute (shared by all 3 layers)
    k_fill1    <<<blk(N, T), T, 0, stream>>>(dinv, N);
    k_deg_count<<<blk(E, T), T, 0, stream>>>(col, dinv, E);
    k_dinv     <<<blk(N, T), T, 0, stream>>>(dinv, N);
    k_norm     <<<blk(E, T), T, 0, stream>>>(row, col, dinv, norm, E);

    const int Mt = (N + 15) / 16;   // 3125, exact

    // ---- layer 1: x(N,128) @ W1(128,64), agg, +b1, relu ----
    k_gemm_wmma <<<Mt * (H / 16), 32, 0, stream>>>(x, W1, lin, N, P, H, H / 16);
    k_agg_init  <<<blk((long long)N * H, T), T, 0, stream>>>(lin, dinv, aggA, N * H, 6);
    k_agg_edges <<<blk((long long)E * H, T), T, 0, stream>>>(row, col, norm, lin, aggA, E, 6);
    k_bias_relu <<<blk((long long)N * H, T), T, 0, stream>>>(aggA, b1, N * H, 6);

    // ---- layer 2: aggA(N,64) @ W2(64,128), agg, +b2, relu ----
    k_gemm_wmma <<<Mt * (2 * H / 16), 32, 0, stream>>>(aggA, W2, lin, N, H, 2 * H, 2 * H / 16);
    k_agg_init  <<<blk((long long)N * 2 * H, T), T, 0, stream>>>(lin, dinv, aggB, N * 2 * H, 7);
    k_agg_edges <<<blk((long long)E * 2 * H, T), T, 0, stream>>>(row, col, norm, lin, aggB, E, 7);
    k_bias_relu <<<blk((long long)N * 2 * H, T), T, 0, stream>>>(aggB, b2, N * 2 * H, 7);

    // ---- layer 3: aggB(N,128) @ W3(128,2), agg, +b3, log_softmax ----
    k_gemm_wmma <<<Mt * 1, 32, 0, stream>>>(aggB, W3, lin, N, 2 * H, C, 1);
    k_agg_init  <<<blk((long long)N * C, T), T, 0, stream>>>(lin, dinv, aggA, N * C, 1);
    k_agg_edges <<<blk((long long)E * C, T), T, 0, stream>>>(row, col, norm, lin, aggA, E, 1);
    k_bias_logsoftmax<<<blk(N, T), T, 0, stream>>>(aggA, b3, (float*)d_out, N);
}